// SimpleRGATLayer_13443247636523
// MI455X (gfx1250) — compile-verified
//
#include <hip/hip_runtime.h>
#include <math.h>

typedef __attribute__((ext_vector_type(2))) float v2f;
typedef __attribute__((ext_vector_type(8))) float v8f;

#define DIN 256
#define HD  128   // H*D = 4*32
#define GEPS 1e-6f

// ---- monotonic float<->uint key for atomic max on signed floats ----
__device__ __forceinline__ unsigned fkey(float f) {
  unsigned b = __float_as_uint(f);
  return b ^ ((unsigned)((int)b >> 31) | 0x80000000u);
}
__device__ __forceinline__ float funkey(unsigned k) {
  unsigned b = (k & 0x80000000u) ? (k ^ 0x80000000u) : ~k;
  return __uint_as_float(b);
}

// ---- init: zero accumulators (d_out used as accumulation buffer) ----
__global__ void init_kernel(float* __restrict__ out, float* __restrict__ denom,
                            unsigned* __restrict__ smax, int N) {
  long i = (long)blockIdx.x * blockDim.x + threadIdx.x;
  long total = (long)N * HD;
  if (i < total) out[i] = 0.0f;
  if (i < (long)N * 4) { denom[i] = 0.0f; smax[i] = 0u; }
}

// ---- h = x @ W via V_WMMA_F32_16X16X4_F32 ----
// One wave computes a 16-row strip across all 128 output cols (8 16x16 tiles).
// A 16x4 f32 frag: lanes 0-15 M=0..15 {v0:K=0, v1:K=1}; lanes 16-31 {v0:K=2, v1:K=3}.
// B 4x16 f32 frag mirrors A: lanes 0-15 {v0:K=0 row, v1:K=1 row}; lanes 16-31 {K=2,K=3}.
// C/D 16x16 f32: VGPR r -> lanes 0-15: M=r, N=lane; lanes 16-31: M=r+8, N=lane-16.
__global__ void gemm_wmma(const float* __restrict__ x, const float* __restrict__ W,
                          float* __restrict__ h, int nRowTiles) {
  int wave = (int)((blockIdx.x * blockDim.x + threadIdx.x) >> 5);
  int lane = threadIdx.x & 31;
  if (wave >= nRowTiles) return;           // whole-wave uniform: EXEC stays all-1s
  const int m0    = wave * 16;
  const int arow  = m0 + (lane & 15);
  const int khalf = lane >> 4;             // 0: K={0,1}, 1: K={2,3}
  const int col   = lane & 15;

  v8f acc[8] = {};                         // 8 col tiles of 16x16 f32
  for (int k = 0; k < DIN; k += 4) {
    const int k0 = k + 2 * khalf;
    v2f a = *(const v2f*)(x + (size_t)arow * DIN + k0);     // contiguous pair
    const float* w0 = W + (size_t)k0 * HD + col;            // row k0
    const float* w1 = w0 + HD;                              // row k0+1
#pragma unroll
    for (int t = 0; t < 8; ++t) {
      v2f b;
      b.x = w0[t * 16];
      b.y = w1[t * 16];
      acc[t] = __builtin_amdgcn_wmma_f32_16x16x4_f32(
          /*neg_a=*/false, a, /*neg_b=*/false, b,
          /*c_mod=*/(short)0, acc[t], /*reuse_a=*/false, /*reuse_b=*/false);
    }
  }
  const int rbase = m0 + khalf * 8;
#pragma unroll
  for (int t = 0; t < 8; ++t)
#pragma unroll
    for (int r = 0; r < 8; ++r)
      h[(size_t)(rbase + r) * HD + t * 16 + col] = acc[t][r];
}

// ---- scalar tail for N % 16 rows (N=50000 -> unused, kept for generality) ----
__global__ void gemm_tail(const float* __restrict__ x, const float* __restrict__ W,
                          float* __restrict__ h, int rowStart, int N) {
  int idx = blockIdx.x * blockDim.x + threadIdx.x;
  int rows = N - rowStart;
  if (idx >= rows * HD) return;
  int r = rowStart + idx / HD, c = idx % HD;
  float s = 0.0f;
  for (int k = 0; k < DIN; ++k) s += x[(size_t)r * DIN + k] * W[(size_t)k * HD + c];
  h[(size_t)r * HD + c] = s;
}

// ---- per-node projections: ps[n,h] = <h[n,h,:], a_src[h,:]>, pd likewise ----
// Factorizes the per-edge dot products: raw[e,h] = ps[src,h] + pd[dst,h].
__global__ void node_proj(const float* __restrict__ h, const float* __restrict__ a_src,
                          const float* __restrict__ a_dst, float* __restrict__ ps,
                          float* __restrict__ pd, int N) {
  int wave = (int)((blockIdx.x * blockDim.x + threadIdx.x) >> 5);
  int lane = threadIdx.x & 31;
  if (wave >= N) return;
  float4 hv = *(const float4*)(h + (size_t)wave * HD + lane * 4);
  float4 as = *(const float4*)(a_src + lane * 4);
  float4 ad = *(const float4*)(a_dst + lane * 4);
  float s = hv.x * as.x + hv.y * as.y + hv.z * as.z + hv.w * as.w;
  float d = hv.x * ad.x + hv.y * ad.y + hv.z * ad.z + hv.w * ad.w;
  // reduce within 8-lane head groups (lane*4 spans one head's 32 dims per 8 lanes)
  s += __shfl_xor(s, 1, 32); d += __shfl_xor(d, 1, 32);
  s += __shfl_xor(s, 2, 32); d += __shfl_xor(d, 2, 32);
  s += __shfl_xor(s, 4, 32); d += __shfl_xor(d, 4, 32);
  if ((lane & 7) == 0) {
    int hh = lane >> 3;
    ps[(size_t)wave * 4 + hh] = s;
    pd[(size_t)wave * 4 + hh] = d;
  }
}

// ---- pass A: scores[e,h] = tanh(ps[src]+pd[dst]) + log(gate); seg-max via atomicMax ----
__global__ void edge_scores(const int* __restrict__ adj, const float* __restrict__ w,
                            const float* __restrict__ ps, const float* __restrict__ pd,
                            float* __restrict__ sc, unsigned* __restrict__ smax, int E) {
  int e = blockIdx.x * blockDim.x + threadIdx.x;
  if (e >= E) return;
  int src = adj[2 * (size_t)e], dst = adj[2 * (size_t)e + 1];
  float g  = fmaxf(w[e], GEPS);
  float lg = logf(g);
  float4 s4 = *(const float4*)(ps + (size_t)src * 4);
  float4 d4 = *(const float4*)(pd + (size_t)dst * 4);
  float r0 = tanhf(s4.x + d4.x) + lg;
  float r1 = tanhf(s4.y + d4.y) + lg;
  float r2 = tanhf(s4.z + d4.z) + lg;
  float r3 = tanhf(s4.w + d4.w) + lg;
  *(float4*)(sc + (size_t)e * 4) = make_float4(r0, r1, r2, r3);
  unsigned* m = smax + (size_t)dst * 4;
  atomicMax(m + 0, fkey(r0));
  atomicMax(m + 1, fkey(r1));
  atomicMax(m + 2, fkey(r2));
  atomicMax(m + 3, fkey(r3));
}

// ---- pass B: ex = exp(score - segmax[dst]); denom[dst] += ex (native f32 atomics) ----
__global__ void edge_exp(const int* __restrict__ adj, float* __restrict__ sc,
                         const unsigned* __restrict__ smax, float* __restrict__ denom, int E) {
  int e = blockIdx.x * blockDim.x + threadIdx.x;
  if (e >= E) return;
  int dst = adj[2 * (size_t)e + 1];
  float4 s = *(const float4*)(sc + (size_t)e * 4);
  const unsigned* m = smax + (size_t)dst * 4;
  float e0 = expf(s.x - funkey(m[0]));
  float e1 = expf(s.y - funkey(m[1]));
  float e2 = expf(s.z - funkey(m[2]));
  float e3 = expf(s.w - funkey(m[3]));
  *(float4*)(sc + (size_t)e * 4) = make_float4(e0, e1, e2, e3);  // overwrite with ex
  float* dn = denom + (size_t)dst * 4;
  __hip_atomic_fetch_add(dn + 0, e0, __ATOMIC_RELAXED, __HIP_MEMORY_SCOPE_AGENT);
  __hip_atomic_fetch_add(dn + 1, e1, __ATOMIC_RELAXED, __HIP_MEMORY_SCOPE_AGENT);
  __hip_atomic_fetch_add(dn + 2, e2, __ATOMIC_RELAXED, __HIP_MEMORY_SCOPE_AGENT);
  __hip_atomic_fetch_add(dn + 3, e3, __ATOMIC_RELAXED, __HIP_MEMORY_SCOPE_AGENT);
}

// ---- pass C: out[dst] += alpha * h[src] ; one wave per edge, 4 elems/lane ----
__global__ void edge_msg(const int* __restrict__ adj, const float* __restrict__ w,
                         const float* __restrict__ ex, const float* __restrict__ denom,
                         const float* __restrict__ h, float* __restrict__ out, int E) {
  int wave = (int)((blockIdx.x * blockDim.x + threadIdx.x) >> 5);
  int lane = threadIdx.x & 31;
  if (wave >= E) return;
  int src = adj[2 * (size_t)wave], dst = adj[2 * (size_t)wave + 1];
  int head = lane >> 3;
  float g = fmaxf(w[wave], GEPS);
  float a = ex[(size_t)wave * 4 + head] / denom[(size_t)dst * 4 + head] * g;
  float4 hv = *(const float4*)(h + (size_t)src * HD + lane * 4);
  float* o = out + (size_t)dst * HD + lane * 4;
  __hip_atomic_fetch_add(o + 0, a * hv.x, __ATOMIC_RELAXED, __HIP_MEMORY_SCOPE_AGENT);
  __hip_atomic_fetch_add(o + 1, a * hv.y, __ATOMIC_RELAXED, __HIP_MEMORY_SCOPE_AGENT);
  __hip_atomic_fetch_add(o + 2, a * hv.z, __ATOMIC_RELAXED, __HIP_MEMORY_SCOPE_AGENT);
  __hip_atomic_fetch_add(o + 3, a * hv.w, __ATOMIC_RELAXED, __HIP_MEMORY_SCOPE_AGENT);
}

// ---- exact GELU in place ----
__global__ void gelu_kernel(float* __restrict__ out, long total) {
  long i = (long)blockIdx.x * blockDim.x + threadIdx.x;
  if (i >= total) return;
  float v = out[i];
  out[i] = 0.5f * v * (1.0f + erff(v * 0.70710678118654752f));
}

extern "C" void kernel_launch(void* const* d_in, const int* in_sizes, int n_in,
                              void* d_out, int out_size, void* d_ws, size_t ws_size,
                              hipStream_t stream) {
  const float* x     = (const float*)d_in[0];
  const int*   adj   = (const int*)d_in[1];
  const float* wts   = (const float*)d_in[2];
  const float* W     = (const float*)d_in[3];
  const float* a_src = (const float*)d_in[4];
  const float* a_dst = (const float*)d_in[5];
  float* out = (float*)d_out;

  const int N = in_sizes[0] / DIN;   // 50000
  const int E = in_sizes[1] / 2;     // 800000

  // workspace carve-up (~42 MB)
  float*    h     = (float*)d_ws;                         // N*128
  float*    ps    = h  + (size_t)N * HD;                  // N*4
  float*    pd    = ps + (size_t)N * 4;                   // N*4
  float*    sc    = pd + (size_t)N * 4;                   // E*4 (scores -> ex)
  unsigned* smax  = (unsigned*)(sc + (size_t)E * 4);      // N*4
  float*    denom = (float*)(smax + (size_t)N * 4);       // N*4

  const long total = (long)N * HD;
  const int  tb = 256;

  init_kernel<<<(int)((total + tb - 1) / tb), tb, 0, stream>>>(out, denom, smax, N);

  const int nRowTiles = N / 16;
  gemm_wmma<<<(nRowTiles + 3) / 4, 4 * 32, 0, stream>>>(x, W, h, nRowTiles);
  const int rem = N - nRowTiles * 16;
  if (rem > 0)
    gemm_tail<<<(rem * HD + tb - 1) / tb, tb, 0, stream>>>(x, W, h, nRowTiles * 16, N);

  node_proj<<<(N + 7) / 8, 256, 0, stream>>>(h, a_src, a_dst, ps, pd, N);
  edge_scores<<<(E + tb - 1) / tb, tb, 0, stream>>>(adj, wts, ps, pd, sc, smax, E);
  edge_exp<<<(E + tb - 1) / tb, tb, 0, stream>>>(adj, sc, smax, denom, E);
  edge_msg<<<(E + 7) / 8, 256, 0, stream>>>(adj, wts, sc, denom, h, out, E);
  gelu_kernel<<<(int)((total + tb - 1) / tb), tb, 0, stream>>>(out, total);
}